// KGPolicy_smodal_58067957842415
// MI455X (gfx1250) — compile-verified
//
#include <hip/hip_runtime.h>
#include <hip/hip_bf16.h>

typedef __attribute__((ext_vector_type(2))) float v2f;
typedef __attribute__((ext_vector_type(8))) float v8f;

#define DFEAT 64
#define NEG_SLOPE 0.2f

// ---------------------------------------------------------------- init
__global__ void gat_init_kernel(float* __restrict__ out, float* __restrict__ amax,
                                float* __restrict__ denom, int nd, int n) {
    int i = blockIdx.x * blockDim.x + threadIdx.x;
    if (i < nd) out[i] = 0.0f;
    if (i < n) {
        amax[i]  = __int_as_float(0xFF800000); // -inf
        denom[i] = 0.0f;
    }
}

// ---------------------------------------------------------------- h = x @ W  (fp32 WMMA 16x16x4)
// One wave computes a 16-row block across all 4 column tiles (D=64).
// A (16x4 fp32) per-lane layout (ISA 7.12.2): lanes 0-15 hold K=k0,k0+1 ; lanes 16-31 hold K=k0+2,k0+3.
// B (4x16 fp32): v0 = rows K=k0 (lanes 0-15) / k0+2 (lanes 16-31); v1 = k0+1 / k0+3.
// C/D (16x16 f32): VGPR j -> M=j (lanes 0-15), M=j+8 (lanes 16-31), N = lane%16.
__global__ void gat_gemm_wmma_kernel(const float* __restrict__ x,
                                     const float* __restrict__ w,
                                     float* __restrict__ h, int nRowBlocks) {
    int wavesPerBlock = blockDim.x >> 5;
    int wave = blockIdx.x * wavesPerBlock + (threadIdx.x >> 5);
    int lane = threadIdx.x & 31;
    if (wave >= nRowBlocks) return;            // wave-uniform: EXEC stays all-ones

    int r0   = wave * 16;
    int mrow = r0 + (lane & 15);
    int koff = (lane >> 4) << 1;               // 0 for lanes 0-15, 2 for lanes 16-31
    int ncol = lane & 15;

    // Preload full 16x64 A panel: 16 k-steps of float2 per lane.
    v2f a[16];
    const float* xrow = x + (size_t)mrow * DFEAT + koff;
#pragma unroll
    for (int ks = 0; ks < 16; ++ks)
        a[ks] = *(const v2f*)(xrow + ks * 4);

#pragma unroll
    for (int ct = 0; ct < 4; ++ct) {
        int c0 = ct * 16;
        v8f acc = {};
#pragma unroll
        for (int ks = 0; ks < 16; ++ks) {
            int k0 = ks * 4;
            v2f b;
            b.x = w[(k0 + koff)     * DFEAT + c0 + ncol];
            b.y = w[(k0 + koff + 1) * DFEAT + c0 + ncol];
            acc = __builtin_amdgcn_wmma_f32_16x16x4_f32(
                false, a[ks], false, b, (short)0, acc, false, false);
        }
        int rbase = r0 + ((lane >> 4) << 3);   // +0 or +8
#pragma unroll
        for (int j = 0; j < 8; ++j)
            h[(size_t)(rbase + j) * DFEAT + c0 + ncol] = acc[j];
    }
}

// ---------------------------------------------------------------- s_i, s_j  (wave per row)
__global__ void gat_score_kernel(const float* __restrict__ h, const float* __restrict__ att,
                                 float* __restrict__ s_i, float* __restrict__ s_j, int n) {
    int wave = blockIdx.x * (blockDim.x >> 5) + (threadIdx.x >> 5);
    int lane = threadIdx.x & 31;
    if (wave >= n) return;
    v2f hv = *(const v2f*)(h + (size_t)wave * DFEAT + lane * 2);
    v2f ai = *(const v2f*)(att + lane * 2);
    v2f aj = *(const v2f*)(att + DFEAT + lane * 2);
    float pi = hv.x * ai.x + hv.y * ai.y;
    float pj = hv.x * aj.x + hv.y * aj.y;
#pragma unroll
    for (int s = 16; s > 0; s >>= 1) {
        pi += __shfl_xor(pi, s, 32);
        pj += __shfl_xor(pj, s, 32);
    }
    if (lane == 0) { s_i[wave] = pi; s_j[wave] = pj; }
}

__device__ __forceinline__ void edge_endpoints(int e, const int* __restrict__ srcA,
                                               const int* __restrict__ dstA,
                                               int ebase, int& s, int& d) {
    if (e < ebase) { s = srcA[e]; d = dstA[e]; }
    else           { s = e - ebase; d = s; }   // self loop
}

__device__ __forceinline__ float leaky(float a) {
    return a > 0.0f ? a : NEG_SLOPE * a;
}

// ---------------------------------------------------------------- segment max (thread per edge)
__global__ void gat_alpha_max_kernel(const int* __restrict__ srcA, const int* __restrict__ dstA,
                                     const float* __restrict__ s_i, const float* __restrict__ s_j,
                                     float* __restrict__ amax, int ebase, int etotal) {
    int e = blockIdx.x * blockDim.x + threadIdx.x;
    if (e >= etotal) return;
    int s, d;
    edge_endpoints(e, srcA, dstA, ebase, s, d);
    float a = leaky(s_i[d] + s_j[s]);
    // float atomic max via int ordering trick (init is -inf = 0xFF800000)
    if (a >= 0.0f) atomicMax((int*)(amax + d), __float_as_int(a));
    else           atomicMin((unsigned int*)(amax + d), __float_as_uint(a));
}

// ---------------------------------------------------------------- segment sum of exp (thread per edge)
__global__ void gat_expsum_kernel(const int* __restrict__ srcA, const int* __restrict__ dstA,
                                  const float* __restrict__ s_i, const float* __restrict__ s_j,
                                  const float* __restrict__ amax, float* __restrict__ denom,
                                  int ebase, int etotal) {
    int e = blockIdx.x * blockDim.x + threadIdx.x;
    if (e >= etotal) return;
    int s, d;
    edge_endpoints(e, srcA, dstA, ebase, s, d);
    float a  = leaky(s_i[d] + s_j[s]);
    float ex = __expf(a - amax[d]);
    unsafeAtomicAdd(denom + d, ex);
}

// ---------------------------------------------------------------- weighted aggregation (wave per edge)
__global__ void gat_aggregate_kernel(const int* __restrict__ srcA, const int* __restrict__ dstA,
                                     const float* __restrict__ s_i, const float* __restrict__ s_j,
                                     const float* __restrict__ amax, const float* __restrict__ denom,
                                     const float* __restrict__ h, float* __restrict__ out,
                                     int ebase, int etotal) {
    int wave = blockIdx.x * (blockDim.x >> 5) + (threadIdx.x >> 5);
    int lane = threadIdx.x & 31;
    if (wave >= etotal) return;
    int s, d;
    edge_endpoints(wave, srcA, dstA, ebase, s, d);
    float wgt = 0.0f;
    if (lane == 0) {
        float a  = leaky(s_i[d] + s_j[s]);
        float ex = __expf(a - amax[d]);
        wgt = ex / (denom[d] + 1e-16f);
    }
    wgt = __shfl(wgt, 0, 32);
    v2f hv = *(const v2f*)(h + (size_t)s * DFEAT + lane * 2);
    float* o = out + (size_t)d * DFEAT + lane * 2;
    unsafeAtomicAdd(o,     hv.x * wgt);
    unsafeAtomicAdd(o + 1, hv.y * wgt);
}

// ---------------------------------------------------------------- bias + L2 normalize (wave per row)
__global__ void gat_normalize_kernel(float* __restrict__ out, const float* __restrict__ bias, int n) {
    int wave = blockIdx.x * (blockDim.x >> 5) + (threadIdx.x >> 5);
    int lane = threadIdx.x & 31;
    if (wave >= n) return;
    float* o = out + (size_t)wave * DFEAT + lane * 2;
    float x0 = o[0] + bias[lane * 2];
    float x1 = o[1] + bias[lane * 2 + 1];
    float ss = x0 * x0 + x1 * x1;
#pragma unroll
    for (int s = 16; s > 0; s >>= 1) ss += __shfl_xor(ss, s, 32);
    float nrm = sqrtf(ss);
    float inv = 1.0f / fmaxf(nrm, 1e-12f);
    o[0] = x0 * inv;
    o[1] = x1 * inv;
}

// ---------------------------------------------------------------- host launcher
extern "C" void kernel_launch(void* const* d_in, const int* in_sizes, int n_in,
                              void* d_out, int out_size, void* d_ws, size_t ws_size,
                              hipStream_t stream) {
    const float* x      = (const float*)d_in[0];
    const int*   eidx   = (const int*)d_in[1];
    const float* weight = (const float*)d_in[2];
    const float* att    = (const float*)d_in[3];
    const float* bias   = (const float*)d_in[4];
    float*       out    = (float*)d_out;

    const int n      = in_sizes[0] / DFEAT;   // 100000
    const int ebase  = in_sizes[1] / 2;       // N*DEG = 3.2M
    const int etotal = ebase + n;             // + self loops
    const int nd     = n * DFEAT;

    const int* srcA = eidx;
    const int* dstA = eidx + ebase;

    // workspace layout (floats): h[n*64] | s_i[n] | s_j[n] | amax[n] | denom[n]
    float* h     = (float*)d_ws;
    float* s_i   = h + (size_t)nd;
    float* s_j   = s_i + n;
    float* amax  = s_j + n;
    float* denom = amax + n;

    const int BLK = 256;               // 8 waves per block (wave32)
    const int WPB = BLK / 32;

    // 1. init out/amax/denom
    gat_init_kernel<<<(nd + BLK - 1) / BLK, BLK, 0, stream>>>(out, amax, denom, nd, n);

    // 2. h = x @ W via fp32 WMMA; one wave per 16-row block
    int rowBlocks = (n + 15) / 16;     // 6250
    gat_gemm_wmma_kernel<<<(rowBlocks + WPB - 1) / WPB, BLK, 0, stream>>>(x, weight, h, rowBlocks);

    // 3. attention scores
    gat_score_kernel<<<(n + WPB - 1) / WPB, BLK, 0, stream>>>(h, att, s_i, s_j, n);

    // 4. segment max of leaky-relu alpha
    gat_alpha_max_kernel<<<(etotal + BLK - 1) / BLK, BLK, 0, stream>>>(
        srcA, dstA, s_i, s_j, amax, ebase, etotal);

    // 5. segment sum of exp
    gat_expsum_kernel<<<(etotal + BLK - 1) / BLK, BLK, 0, stream>>>(
        srcA, dstA, s_i, s_j, amax, denom, ebase, etotal);

    // 6. weighted scatter aggregation (wave per edge)
    gat_aggregate_kernel<<<(etotal + WPB - 1) / WPB, BLK, 0, stream>>>(
        srcA, dstA, s_i, s_j, amax, denom, h, out, ebase, etotal);

    // 7. bias + row L2 normalize in place
    gat_normalize_kernel<<<(n + WPB - 1) / WPB, BLK, 0, stream>>>(out, bias, n);
}